// SpikeFP32Exp_43860206027311
// MI455X (gfx1250) — compile-verified
//
#include <hip/hip_runtime.h>
#include <stdint.h>

// ---------------------------------------------------------------------------
// SpikeFP32Exp decode: x[...,32] bits of an fp32 encoding -> 5-bit fixed-point
//   E     = bits 1..8 (MSB first), shift = (E + 129) & 255   (= E - 127)
//   V     = 0b1 m0 m1 m2 m3      (bits 9..12)
//   O     = shift <= 4 ? V >> (4 - shift) : 0
//   out[c]= bit (4 - c) of O, as float 0/1
// HBM-bound; input staged with the Tensor Data Mover reading only the first
// 64B of each 128B record (2D tile: 16 dwords x 256 rows, stride 32 dwords),
// with DMA padding 16dw+4dw -> 80B LDS row stride to reduce bank conflicts.
// ---------------------------------------------------------------------------

typedef __attribute__((ext_vector_type(4))) unsigned int tdm_v4u;
typedef __attribute__((ext_vector_type(8))) int          tdm_v8i;
typedef __attribute__((ext_vector_type(4))) int          tdm_v4i;

#define TILE_ROWS     256
#define ROW_LDS_BYTES 80   // 16 data dwords + 4 pad dwords

__device__ __forceinline__ void tdm_load_tile(const float* gaddr,
                                              unsigned ldsOff,
                                              unsigned rows) {
    unsigned long long ga = (unsigned long long)(uintptr_t)gaddr;

    // D# group 0: count=1 (valid user descriptor), lds_addr, 57b global, type=2
    tdm_v4u g0;
    g0[0] = 1u;
    g0[1] = ldsOff;
    g0[2] = (unsigned)ga;
    g0[3] = ((unsigned)(ga >> 32) & 0x01FFFFFFu) | (2u << 30);

    // D# group 1
    const unsigned d0 = (2u << 16)   // data_size = 4 bytes
                      | (1u << 20)   // pad_enable
                      | (3u << 22)   // pad_interval: (1<<3)*8B = 64B = 16 dwords
                      | (3u << 25);  // pad_amount : 4 dwords (16B)
    const unsigned tdim0   = 16u;    // tensor length dim0 (dwords we care about)
    const unsigned tile0   = 16u;    // tile dim0 = 16 dwords = 64B per record
    const unsigned stride0 = 32u;    // 128B record stride, in dword units
    tdm_v8i g1;
    g1[0] = (int)d0;
    g1[1] = (int)((tdim0 & 0xFFFFu) << 16);                              // abar=0 | tdim0.lo
    g1[2] = (int)(((tdim0 >> 16) & 0xFFFFu) | ((rows & 0xFFFFu) << 16)); // tdim0.hi | tdim1.lo
    g1[3] = (int)(((rows >> 16) & 0xFFFFu) | ((tile0 & 0xFFFFu) << 16)); // tdim1.hi | tile_dim0
    g1[4] = (int)(rows & 0xFFFFu);                                       // tile_dim1 | tile_dim2=0
    g1[5] = (int)stride0;                                                // tensor_dim0_stride.lo
    g1[6] = 0;                                                           // stride0.hi | stride1.lo
    g1[7] = 0;                                                           // stride1.hi

    tdm_v4i g2 = {0, 0, 0, 0};               // group 2: unused (2-D tile)
    tdm_v4i g3 = {0, 0, 0, 0};               // group 3: unused
    tdm_v8i g4 = {0, 0, 0, 0, 0, 0, 0, 0};   // extra group (clang-23 6-arg form)

    __builtin_amdgcn_tensor_load_to_lds(g0, g1, g2, g3, g4, 0);
}

__global__ void __launch_bounds__(256)
spike_decode_kernel(const float* __restrict__ x, float* __restrict__ out,
                    int Npix, int numTiles) {
    __shared__ __align__(16) unsigned char lbuf[2][TILE_ROWS * ROW_LDS_BYTES];

    const int  tid     = (int)threadIdx.x;
    const bool isWave0 = (tid >> 5) == 0;
    // Flat shared-aperture address: low 32 bits are the LDS byte offset.
    const unsigned ldsOff0 = (unsigned)(uintptr_t)(void*)&lbuf[0][0];
    const unsigned ldsOff1 = (unsigned)(uintptr_t)(void*)&lbuf[1][0];

    int t   = (int)blockIdx.x;
    int cur = 0;

    if (t < numTiles && isWave0) {
        int rows = Npix - t * TILE_ROWS;
        if (rows > TILE_ROWS) rows = TILE_ROWS;
        tdm_load_tile(x + (size_t)t * TILE_ROWS * 32, ldsOff0, (unsigned)rows);
    }

    for (; t < numTiles; t += (int)gridDim.x, cur ^= 1) {
        int tn = t + (int)gridDim.x;
        if (isWave0) {
            if (tn < numTiles) {
                int rows = Npix - tn * TILE_ROWS;
                if (rows > TILE_ROWS) rows = TILE_ROWS;
                tdm_load_tile(x + (size_t)tn * TILE_ROWS * 32,
                              cur ? ldsOff0 : ldsOff1, (unsigned)rows);
                __builtin_amdgcn_s_wait_tensorcnt(1); // tile t's DMA complete
            } else {
                __builtin_amdgcn_s_wait_tensorcnt(0);
            }
        }
        __syncthreads();

        size_t pix = (size_t)t * TILE_ROWS + (size_t)tid;
        if (pix < (size_t)Npix) {
            const uint4* r4 =
                (const uint4*)(&lbuf[cur][0] + (size_t)tid * ROW_LDS_BYTES);
            uint4 w0 = r4[0];   // dwords 0..3   (bit0 = sign, unused)
            uint4 w1 = r4[1];   // dwords 4..7
            uint4 w2 = r4[2];   // dwords 8..11
            uint4 w3 = r4[3];   // dwords 12..15

            unsigned E = ((unsigned)(w0.y != 0u) << 7) |
                         ((unsigned)(w0.z != 0u) << 6) |
                         ((unsigned)(w0.w != 0u) << 5) |
                         ((unsigned)(w1.x != 0u) << 4) |
                         ((unsigned)(w1.y != 0u) << 3) |
                         ((unsigned)(w1.z != 0u) << 2) |
                         ((unsigned)(w1.w != 0u) << 1) |
                         ((unsigned)(w2.x != 0u) << 0);
            unsigned shift = (E + 129u) & 255u;
            unsigned V = 16u |
                         ((unsigned)(w2.y != 0u) << 3) |
                         ((unsigned)(w2.z != 0u) << 2) |
                         ((unsigned)(w2.w != 0u) << 1) |
                         ((unsigned)(w3.x != 0u) << 0);
            unsigned O = (shift <= 4u) ? (V >> (4u - shift)) : 0u;

            float* o = out + pix * 5u;
            o[0] = (float)((O >> 4) & 1u);
            o[1] = (float)((O >> 3) & 1u);
            o[2] = (float)((O >> 2) & 1u);
            o[3] = (float)((O >> 1) & 1u);
            o[4] = (float)(O & 1u);
        }
        __syncthreads();  // all reads of lbuf[cur] done before it is re-filled
    }
}

extern "C" void kernel_launch(void* const* d_in, const int* in_sizes, int n_in,
                              void* d_out, int out_size, void* d_ws, size_t ws_size,
                              hipStream_t stream) {
    const float* x   = (const float*)d_in[0];
    float*       out = (float*)d_out;

    int Npix     = in_sizes[0] / 32;                       // 1M pixels
    int numTiles = (Npix + TILE_ROWS - 1) / TILE_ROWS;     // 4096
    int blocks   = numTiles < 2048 ? numTiles : 2048;      // 2 tiles/block -> DB overlap

    hipLaunchKernelGGL(spike_decode_kernel, dim3(blocks), dim3(256), 0, stream,
                       x, out, Npix, numTiles);
}